// SinkhornDistance_1D_75471165325372
// MI455X (gfx1250) — compile-verified
//
#include <hip/hip_runtime.h>

typedef float v2f __attribute__((ext_vector_type(2)));
typedef float v8f __attribute__((ext_vector_type(8)));

#define NPT        1024
#define BATCH      8
#define WPB        8               // workgroups per batch
#define NWG        (BATCH * WPB)   // 64 total workgroups
#define TPB        256             // 8 wave32 per workgroup
#define ROWS_PER_WG (NPT / WPB)    // 128
#define MAX_ITER   20

// constants (eps = 0.1)
#define IE2_C   14.426950408889634f   // (1/eps) * log2(e)
#define LN2_C   0.69314718055994531f
#define EPS_C   0.1f
#define LOGMU_C (-6.9314616f)         // log(1/1024 + 1e-8)

__device__ __forceinline__ void grid_barrier(int* cnt, int slot) {
    __syncthreads();
    if (threadIdx.x == 0) {
        __threadfence();
        __hip_atomic_fetch_add(&cnt[slot], 1, __ATOMIC_ACQ_REL, __HIP_MEMORY_SCOPE_AGENT);
        while (__hip_atomic_load(&cnt[slot], __ATOMIC_ACQUIRE, __HIP_MEMORY_SCOPE_AGENT) < NWG)
            __builtin_amdgcn_s_sleep(1);
    }
    __syncthreads();
}

// Row logsumexp (base-2 domain) over 1024 columns for the 16 rows owned by this
// wave. Lane layout: row = lane&15 (both halves), lane half h covers columns
// {jc+2h, jc+2h+1} per 4-wide chunk. Row sums are produced by
// v_wmma_f32_16x16x4_f32 with A = ones(16x4): D[m,n] = sum_k B[k,n], so every
// lane's acc[0] holds its own row's sum (both lane halves combined by the MMA).
__device__ __forceinline__ float tile_lse(const float2* __restrict__ sP,
                                          float aP, float bt, int h) {
    float m0 = -3.0e38f, m1 = -3.0e38f;
    #pragma unroll 4
    for (int jc = 0; jc < NPT; jc += 4) {
        float2 q0 = sP[jc + 2 * h];
        float2 q1 = sP[jc + 2 * h + 1];
        m0 = fmaxf(m0, fmaf(bt, q0.x, aP + q0.y));
        m1 = fmaxf(m1, fmaf(bt, q1.x, aP + q1.y));
    }
    float m2 = fmaxf(m0, m1);
    m2 = fmaxf(m2, __shfl_xor(m2, 16, 32));   // combine the two j-halves of the row

    const float am = aP - m2;
    v8f acc = {0.f, 0.f, 0.f, 0.f, 0.f, 0.f, 0.f, 0.f};
    v2f ones = {1.0f, 1.0f};
    #pragma unroll 4
    for (int jc = 0; jc < NPT; jc += 4) {
        float2 q0 = sP[jc + 2 * h];
        float2 q1 = sP[jc + 2 * h + 1];
        v2f e;
        e.x = __builtin_amdgcn_exp2f(fmaf(bt, q0.x, am + q0.y));
        e.y = __builtin_amdgcn_exp2f(fmaf(bt, q1.x, am + q1.y));
        acc = __builtin_amdgcn_wmma_f32_16x16x4_f32(false, ones, false, e,
                                                    (short)0, acc, false, false);
    }
    return LN2_C * (m2 + __builtin_amdgcn_logf(acc[0]));   // v_log_f32 = log2
}

__global__ void __launch_bounds__(TPB)
sinkhorn_kernel(const float* __restrict__ x, const float* __restrict__ y,
                float* __restrict__ out, float* U, float* V,
                float* ERR, int* CNT) {
    __shared__ float  sX[NPT];
    __shared__ float  sY[NPT];
    __shared__ float2 sP[NPT];     // {coord[j], (dual[j] - coord[j]^2) * IE2}
    __shared__ float  sErr;

    const int b    = blockIdx.x / WPB;
    const int wb   = blockIdx.x % WPB;
    const int tid  = threadIdx.x;
    const int lane = tid & 31;
    const int wave = tid >> 5;
    const int h    = lane >> 4;
    const int r    = wb * ROWS_PER_WG + wave * 16 + (lane & 15);

    const float* xb = x + b * NPT;
    const float* yb = y + b * NPT;
    float* Ub = U + b * NPT;
    float* Vb = V + b * NPT;

    for (int i = tid; i < NPT; i += TPB) { sX[i] = xb[i]; sY[i] = yb[i]; }
    __syncthreads();

    const float x_r = sX[r];
    const float y_r = sY[r];

    for (int it = 0; it < MAX_ITER; ++it) {
        // ---------------- u update: lse over columns j ----------------
        if (tid == 0) sErr = 0.0f;
        for (int i = tid; i < NPT; i += TPB) {
            float vj = Vb[i];
            float yj = sY[i];
            sP[i] = make_float2(yj, (vj - yj * yj) * IE2_C);
        }
        __syncthreads();

        float u_r = Ub[r];
        float aP  = (u_r - x_r * x_r) * IE2_C;
        float bt  = 2.0f * x_r * IE2_C;
        float lse = tile_lse(sP, aP, bt, h);
        float un  = EPS_C * (LOGMU_C - lse) + u_r;
        float dl  = fabsf(un - u_r);
        if (h == 0) Ub[r] = un; else dl = 0.0f;
        #pragma unroll
        for (int off = 16; off >= 1; off >>= 1) dl += __shfl_xor(dl, off, 32);
        if (lane == 0) atomicAdd(&sErr, dl);
        __syncthreads();
        if (tid == 0) atomicAdd(&ERR[it], sErr);
        grid_barrier(CNT, 2 * it);

        // ---------------- v update: lse over rows i --------------------
        for (int i = tid; i < NPT; i += TPB) {
            float ui = Ub[i];
            float xi = sX[i];
            sP[i] = make_float2(xi, (ui - xi * xi) * IE2_C);
        }
        __syncthreads();
        float v_c  = Vb[r];
        float aPv  = (v_c - y_r * y_r) * IE2_C;
        float btv  = 2.0f * y_r * IE2_C;
        float lsev = tile_lse(sP, aPv, btv, h);
        float vn   = EPS_C * (LOGMU_C - lsev) + v_c;
        if (h == 0) Vb[r] = vn;
        grid_barrier(CNT, 2 * it + 1);

        float errsum = __hip_atomic_load(&ERR[it], __ATOMIC_RELAXED,
                                         __HIP_MEMORY_SCOPE_AGENT);
        if (errsum < 0.8f) break;   // mean_b(sum_i |du|) < 0.1  (uniform across grid)
    }

    // ---------------- cost[b,i] = sum_j exp(M) * C -----------------
    for (int i = tid; i < NPT; i += TPB) {
        float vj = Vb[i];
        float yj = sY[i];
        sP[i] = make_float2(yj, (vj - yj * yj) * IE2_C);
    }
    __syncthreads();
    float u_r = Ub[r];
    float aP  = (u_r - x_r * x_r) * IE2_C;
    float bt  = 2.0f * x_r * IE2_C;
    v8f acc = {0.f, 0.f, 0.f, 0.f, 0.f, 0.f, 0.f, 0.f};
    v2f ones = {1.0f, 1.0f};
    #pragma unroll 4
    for (int jc = 0; jc < NPT; jc += 4) {
        float2 q0 = sP[jc + 2 * h];
        float2 q1 = sP[jc + 2 * h + 1];
        float p0 = __builtin_amdgcn_exp2f(fmaf(bt, q0.x, aP + q0.y));
        float p1 = __builtin_amdgcn_exp2f(fmaf(bt, q1.x, aP + q1.y));
        float d0 = x_r - q0.x;
        float d1 = x_r - q1.x;
        v2f pc = {p0 * d0 * d0, p1 * d1 * d1};
        acc = __builtin_amdgcn_wmma_f32_16x16x4_f32(false, ones, false, pc,
                                                    (short)0, acc, false, false);
    }
    if (h == 0) out[b * NPT + r] = acc[0];
}

extern "C" void kernel_launch(void* const* d_in, const int* in_sizes, int n_in,
                              void* d_out, int out_size, void* d_ws, size_t ws_size,
                              hipStream_t stream) {
    (void)in_sizes; (void)n_in; (void)out_size; (void)ws_size;
    const float* x = (const float*)d_in[0];
    const float* y = (const float*)d_in[1];
    float* out = (float*)d_out;

    float* U   = (float*)d_ws;            // 8192 floats
    float* V   = U + BATCH * NPT;         // 8192 floats
    float* ERR = V + BATCH * NPT;         // 20 floats (padded to 32)
    int*   CNT = (int*)(ERR + 32);        // 40 barrier slots (padded to 64)

    // zero u, v (reference inits duals to 0), err accumulators, barrier counters
    size_t clear_bytes = (size_t)(2 * BATCH * NPT + 32 + 64) * sizeof(float);
    hipMemsetAsync(d_ws, 0, clear_bytes, stream);

    sinkhorn_kernel<<<dim3(NWG), dim3(TPB), 0, stream>>>(x, y, out, U, V, ERR, CNT);
}